// Discriminator_3135326126243
// MI455X (gfx1250) — compile-verified
//
#include <hip/hip_runtime.h>
#include <cmath>

typedef __attribute__((ext_vector_type(2))) float v2f;
typedef __attribute__((ext_vector_type(8))) float v8f;

constexpr int H   = 40;    // hidden size
constexpr int N3  = 120;   // 3*H gate width
constexpr int HK  = 10;    // 40/4 K-steps for hidden contribution

// CDNA5 fp32 tensor op: D(16x16) = A(16x4) * B(4x16) + C
__device__ __forceinline__ v8f wmma4(v2f a, v2f b, v8f c) {
  return __builtin_amdgcn_wmma_f32_16x16x4_f32(
      /*neg_a=*/false, a, /*neg_b=*/false, b,
      /*c_mod=*/(short)0, c, /*reuse_a=*/false, /*reuse_b=*/false);
}

// Cross-lane LDS fence: wave32 lanes exchange through LDS; compiler only
// tracks per-thread DS dependencies, so fence explicitly.
__device__ __forceinline__ void dswait() {
  asm volatile("s_wait_dscnt 0" ::: "memory");
}

__device__ __forceinline__ float sigm(float x) { return 1.0f / (1.0f + __expf(-x)); }

// ---------------------------------------------------------------------------
// Fused GRU layer: input projection + recurrent projection + gates, scanned
// over T by persistent waves. One wave (32 threads) per block owns 16 batch
// rows. D = input feature dim (20 for layer 0, 40 otherwise).
// act: [B,T,D], wih: [120,D], whh: [120,40], out: [B,T,40]
// ---------------------------------------------------------------------------
template <int D>
__global__ __launch_bounds__(32)
void gru_scan_kernel(const float* __restrict__ act,
                     const float* __restrict__ wih,
                     const float* __restrict__ whh,
                     const float* __restrict__ bih,
                     const float* __restrict__ bhh,
                     float* __restrict__ out,
                     int B, int T) {
  constexpr int DK   = D / 4;       // K-steps for input contribution
  constexpr int KC   = HK + DK;     // combined K-steps for r/z gates
  constexpr int SSTR = 216;         // staging stride (even -> b64-aligned cols)

  // Weight fragments packed in lane-major WMMA B-fragment order.
  __shared__ __align__(16) float wRZ[5 * KC * 64];  // cols 0..79, K=[h|x]
  __shared__ __align__(16) float wHN[3 * HK * 64];  // cols 80..119, K=h
  __shared__ __align__(16) float wXN[3 * DK * 64];  // cols 80..119, K=x
  __shared__ __align__(16) float biasS[240];        // [bih | bhh]
  // Per-wave staging: [0,80)=rz, [80,128)=hn, [128,176)=xn, [176,216)=h
  __shared__ __align__(16) float stage[16 * SSTR];

  const int lane    = threadIdx.x & 31;
  const int rowBase = blockIdx.x * 16;
  if (rowBase >= B) return;
  const int mrow  = lane & 15;         // M row (A frag) or N col (B/C frag)
  const int khalf = (lane >> 4) * 2;   // K offset for upper lane half

  // ---- one-time: pack weights into B-fragment layout in LDS ----
  for (int nt = 0; nt < 5; ++nt)
    for (int ks = 0; ks < KC; ++ks) {
      int n  = nt * 16 + mrow;
      int k0 = ks * 4 + khalf;
      float* p = &wRZ[((nt * KC + ks) * 32 + lane) * 2];
      for (int j = 0; j < 2; ++j) {
        int k = k0 + j;
        p[j] = (k < H) ? whh[n * H + k] : wih[n * D + (k - H)];
      }
    }
  for (int nt = 0; nt < 3; ++nt) {
    int n = 80 + nt * 16 + mrow;       // pad cols >= 120 with zeros
    for (int ks = 0; ks < HK; ++ks) {
      int k0 = ks * 4 + khalf;
      float* p = &wHN[((nt * HK + ks) * 32 + lane) * 2];
      p[0] = (n < N3) ? whh[n * H + k0]     : 0.0f;
      p[1] = (n < N3) ? whh[n * H + k0 + 1] : 0.0f;
    }
    for (int ks = 0; ks < DK; ++ks) {
      int k0 = ks * 4 + khalf;
      float* p = &wXN[((nt * DK + ks) * 32 + lane) * 2];
      p[0] = (n < N3) ? wih[n * D + k0]     : 0.0f;
      p[1] = (n < N3) ? wih[n * D + k0 + 1] : 0.0f;
    }
  }
  for (int i = lane; i < N3; i += 32) {
    biasS[i]       = bih[i];
    biasS[120 + i] = bhh[i];
  }

  // ---- h = 0 ----
  v2f ah[HK];
  #pragma unroll
  for (int ks = 0; ks < HK; ++ks) ah[ks] = (v2f){0.0f, 0.0f};
  for (int i = 0; i < 20; ++i) {
    int idx = lane + 32 * i;
    stage[(idx / H) * SSTR + 176 + (idx % H)] = 0.0f;
  }
  dswait();

  for (int t = 0; t < T; ++t) {
    // ---- load x A-fragments for this timestep (aligned b64 per lane) ----
    v2f ax[DK];
    const float* arow = act + ((size_t)(rowBase + mrow) * T + t) * D;
    #pragma unroll
    for (int ks = 0; ks < DK; ++ks)
      ax[ks] = *(const v2f*)(arow + ks * 4 + khalf);

    // ---- r/z gates: cols 0..79, fused K = [h(40) | x(D)] ----
    #pragma unroll
    for (int nt = 0; nt < 5; ++nt) {
      v8f acc = {0, 0, 0, 0, 0, 0, 0, 0};
      #pragma unroll
      for (int ks = 0; ks < HK; ++ks)
        acc = wmma4(ah[ks], *(const v2f*)&wRZ[((nt * KC + ks) * 32 + lane) * 2], acc);
      #pragma unroll
      for (int ks = 0; ks < DK; ++ks)
        acc = wmma4(ax[ks], *(const v2f*)&wRZ[((nt * KC + HK + ks) * 32 + lane) * 2], acc);
      #pragma unroll
      for (int r = 0; r < 8; ++r)
        stage[(r + khalf * 4) * SSTR + nt * 16 + mrow] = acc[r];
    }
    // ---- hn: cols 80..119, h contribution only ----
    #pragma unroll
    for (int nt = 0; nt < 3; ++nt) {
      v8f acc = {0, 0, 0, 0, 0, 0, 0, 0};
      #pragma unroll
      for (int ks = 0; ks < HK; ++ks)
        acc = wmma4(ah[ks], *(const v2f*)&wHN[((nt * HK + ks) * 32 + lane) * 2], acc);
      #pragma unroll
      for (int r = 0; r < 8; ++r)
        stage[(r + khalf * 4) * SSTR + 80 + nt * 16 + mrow] = acc[r];
    }
    // ---- xn: cols 80..119, x contribution only ----
    #pragma unroll
    for (int nt = 0; nt < 3; ++nt) {
      v8f acc = {0, 0, 0, 0, 0, 0, 0, 0};
      #pragma unroll
      for (int ks = 0; ks < DK; ++ks)
        acc = wmma4(ax[ks], *(const v2f*)&wXN[((nt * DK + ks) * 32 + lane) * 2], acc);
      #pragma unroll
      for (int r = 0; r < 8; ++r)
        stage[(r + khalf * 4) * SSTR + 128 + nt * 16 + mrow] = acc[r];
    }
    dswait();

    // ---- gates: 640 (row,col) cells, 20 per lane ----
    #pragma unroll
    for (int i = 0; i < 20; ++i) {
      int idx = lane + 32 * i;
      int r = idx / H, c = idx - r * H;
      const float* sr = &stage[r * SSTR];
      float rv = sigm(sr[c]      + biasS[c]      + biasS[120 + c]);
      float zv = sigm(sr[40 + c] + biasS[40 + c] + biasS[160 + c]);
      float hn = sr[80 + c]  + biasS[200 + c];   // W_hn h + b_hn
      float xn = sr[128 + c] + biasS[80 + c];    // W_in x + b_in
      float nv = tanhf(xn + rv * hn);
      float hv = (1.0f - zv) * nv + zv * sr[176 + c];
      stage[r * SSTR + 176 + c] = hv;
      out[((size_t)(rowBase + r) * T + t) * H + c] = hv;
    }
    dswait();

    // ---- rebuild h A-fragments for next timestep ----
    #pragma unroll
    for (int ks = 0; ks < HK; ++ks)
      ah[ks] = *(const v2f*)&stage[mrow * SSTR + 176 + ks * 4 + khalf];
  }
}

// ---------------------------------------------------------------------------
// Fused 2-layer MLP: out = leaky(in@w1^T + b1)@w2^T + b2 over nRows rows.
// 4 waves per block; each wave computes one 16-row tile with WMMA.
// ---------------------------------------------------------------------------
__global__ __launch_bounds__(128)
void mlp_kernel(const float* __restrict__ in,
                const float* __restrict__ w1, const float* __restrict__ b1,
                const float* __restrict__ w2, const float* __restrict__ b2,
                float* __restrict__ out, int nRows) {
  __shared__ __align__(16) float w1p[3 * HK * 64];
  __shared__ __align__(16) float w2p[3 * HK * 64];
  __shared__ __align__(16) float b1S[48], b2S[48];
  __shared__ __align__(16) float stageW[4 * 16 * 48];

  const int lane  = threadIdx.x & 31;
  const int wave  = threadIdx.x >> 5;
  const int mrow  = lane & 15;
  const int khalf = (lane >> 4) * 2;

  for (int w = wave; w < 6; w += 4) {      // 2 matrices x 3 N-tiles
    const float* W = (w < 3) ? w1 : w2;
    float*       P = (w < 3) ? w1p : w2p;
    int nt = (w < 3) ? w : w - 3;
    int n  = nt * 16 + mrow;               // pad cols 40..47 with zero
    for (int ks = 0; ks < HK; ++ks) {
      int k0 = ks * 4 + khalf;
      float* p = &P[((nt * HK + ks) * 32 + lane) * 2];
      p[0] = (n < H) ? W[n * H + k0]     : 0.0f;
      p[1] = (n < H) ? W[n * H + k0 + 1] : 0.0f;
    }
  }
  if (threadIdx.x < 48) {
    b1S[threadIdx.x] = (threadIdx.x < H) ? b1[threadIdx.x] : 0.0f;
    b2S[threadIdx.x] = (threadIdx.x < H) ? b2[threadIdx.x] : 0.0f;
  }
  __syncthreads();

  const int tile = blockIdx.x * 4 + wave;
  if (tile * 16 >= nRows) return;
  const int rowBase = tile * 16;
  float* st = &stageW[wave * 16 * 48];

  v2f ax[HK];
  const float* arow = in + (size_t)(rowBase + mrow) * H;
  #pragma unroll
  for (int ks = 0; ks < HK; ++ks) ax[ks] = *(const v2f*)(arow + ks * 4 + khalf);

  #pragma unroll
  for (int nt = 0; nt < 3; ++nt) {
    v8f acc = {0, 0, 0, 0, 0, 0, 0, 0};
    #pragma unroll
    for (int ks = 0; ks < HK; ++ks)
      acc = wmma4(ax[ks], *(const v2f*)&w1p[((nt * HK + ks) * 32 + lane) * 2], acc);
    int col = nt * 16 + mrow;
    #pragma unroll
    for (int r = 0; r < 8; ++r) {
      float v = acc[r] + b1S[col];
      v = (v > 0.0f) ? v : 0.01f * v;      // leaky relu
      st[(r + khalf * 4) * 48 + col] = v;
    }
  }
  dswait();
  v2f aa[HK];
  #pragma unroll
  for (int ks = 0; ks < HK; ++ks)
    aa[ks] = *(const v2f*)&st[mrow * 48 + ks * 4 + khalf];

  #pragma unroll
  for (int nt = 0; nt < 3; ++nt) {
    v8f acc = {0, 0, 0, 0, 0, 0, 0, 0};
    #pragma unroll
    for (int ks = 0; ks < HK; ++ks)
      acc = wmma4(aa[ks], *(const v2f*)&w2p[((nt * HK + ks) * 32 + lane) * 2], acc);
    int col = nt * 16 + mrow;
    if (col < H) {
      #pragma unroll
      for (int r = 0; r < 8; ++r)
        out[(size_t)(rowBase + r + khalf * 4) * H + col] = acc[r] + b2S[col];
    }
  }
}

// ---------------------------------------------------------------------------
// Head: sigmoid(w_last . leaky(w1 h_{T-1} + b1) + b_last) per batch row.
// ---------------------------------------------------------------------------
__global__ void head_kernel(const float* __restrict__ hseq,
                            const float* __restrict__ w1, const float* __restrict__ b1,
                            const float* __restrict__ wl, const float* __restrict__ bl,
                            float* __restrict__ outp, int B, int T) {
  int b = blockIdx.x * blockDim.x + threadIdx.x;
  if (b >= B) return;
  const float* hp = hseq + ((size_t)b * T + (T - 1)) * H;
  float hv[H];
  #pragma unroll
  for (int k = 0; k < H; ++k) hv[k] = hp[k];
  float acc = bl[0];
  for (int j = 0; j < H; ++j) {
    float s = b1[j];
    #pragma unroll
    for (int k = 0; k < H; ++k) s += w1[j * H + k] * hv[k];
    s = (s > 0.0f) ? s : 0.01f * s;
    acc += wl[j] * s;
  }
  outp[b] = 1.0f / (1.0f + __expf(-acc));
}

extern "C" void kernel_launch(void* const* d_in, const int* in_sizes, int n_in,
                              void* d_out, int out_size, void* d_ws, size_t ws_size,
                              hipStream_t stream) {
  (void)n_in; (void)out_size; (void)ws_size;
  const float* x      = (const float*)d_in[0];
  const float* g0_wih = (const float*)d_in[1];
  const float* g0_whh = (const float*)d_in[2];
  const float* g0_bih = (const float*)d_in[3];
  const float* g0_bhh = (const float*)d_in[4];
  const float* g_wih  = (const float*)d_in[5];
  const float* g_whh  = (const float*)d_in[6];
  const float* g_bih  = (const float*)d_in[7];
  const float* g_bhh  = (const float*)d_in[8];
  const float* mlp_w1 = (const float*)d_in[9];
  const float* mlp_b1 = (const float*)d_in[10];
  const float* mlp_w2 = (const float*)d_in[11];
  const float* mlp_b2 = (const float*)d_in[12];
  const float* w_last = (const float*)d_in[13];
  const float* b_last = (const float*)d_in[14];

  const int T = 512;
  const int B = in_sizes[0] / (T * 20);
  float* bufA = (float*)d_ws;                     // [B,T,40]
  float* bufB = bufA + (size_t)B * T * H;         // [B,T,40]

  const int scanGrid = B / 16;                    // one 16-row wave per block
  const int nRows    = B * T;
  const int mlpGrid  = nRows / 64;                // 4 tiles per 128-thread block

  // lay1: 2-layer GRU
  gru_scan_kernel<20><<<scanGrid, 32, 0, stream>>>(x, g0_wih, g0_whh, g0_bih, g0_bhh,
                                                   bufA, B, T);
  gru_scan_kernel<40><<<scanGrid, 32, 0, stream>>>(bufA, g_wih, g_whh, g_bih, g_bhh,
                                                   bufB, B, T);
  // lay2..lay7: (MLP, GRU) x 3 with ping-pong buffers
  for (int i = 0; i < 3; ++i) {
    mlp_kernel<<<mlpGrid, 128, 0, stream>>>(bufB, mlp_w1 + i * H * H, mlp_b1 + i * H,
                                            mlp_w2 + i * H * H, mlp_b2 + i * H,
                                            bufA, nRows);
    gru_scan_kernel<40><<<scanGrid, 32, 0, stream>>>(
        bufA, g_wih + (size_t)(i + 1) * N3 * H, g_whh + (size_t)(i + 1) * N3 * H,
        g_bih + (size_t)(i + 1) * N3, g_bhh + (size_t)(i + 1) * N3, bufB, B, T);
  }
  // lay8 head on last timestep only
  head_kernel<<<(B + 255) / 256, 256, 0, stream>>>(bufB, mlp_w1 + 3 * H * H,
                                                   mlp_b1 + 3 * H, w_last, b_last,
                                                   (float*)d_out, B, T);
}